// Net_10969346474784
// MI455X (gfx1250) — compile-verified
//
#include <hip/hip_runtime.h>
#include <math.h>

// ---------------------------------------------------------------------------
// GCN (2x GCNConv + scatter_mean + linear + log_softmax) for MI455X / gfx1250.
// f32 end-to-end; GEMMs use V_WMMA_F32_16X16X4_F32 (wave32 WMMA).
// Working set (~154MB) fits in the 192MB L2, so edge gather/scatter is
// L2-resident; only first touch of x (102MB) hits HBM (~5us @ 23.3TB/s).
// ---------------------------------------------------------------------------

typedef __attribute__((ext_vector_type(2))) float v2f;
typedef __attribute__((ext_vector_type(8))) float v8f;

// ---------------- utility fills -------------------------------------------
__global__ void k_fill(float* __restrict__ p, float v, size_t n) {
  size_t i = (size_t)blockIdx.x * blockDim.x + threadIdx.x;
  size_t st = (size_t)gridDim.x * blockDim.x;
  for (; i < n; i += st) p[i] = v;
}

// ---------------- degree / dinv -------------------------------------------
__global__ void k_degree(const int* __restrict__ dst, float* __restrict__ deg, int nE) {
  int e = blockIdx.x * blockDim.x + threadIdx.x;
  if (e < nE) atomicAdd(deg + dst[e], 1.0f);
}

__global__ void k_rsqrt(float* __restrict__ p, int n) {
  int i = blockIdx.x * blockDim.x + threadIdx.x;
  if (i < n) p[i] = rsqrtf(p[i]);   // deg >= 1 always (self loops), so no guard needed
}

// ---------------- WMMA f32 GEMM: C[nrows,ND] = A[nrows,KD] @ B[KD,ND] ------
// 256 threads = 8 wave32; each wave computes one 16x16 tile with
// V_WMMA_F32_16X16X4_F32 over KD/4 steps.
// A fragment (16x4 f32): lanes 0-15 row=lane K={0,1}; lanes 16-31 K={2,3}.
// B fragment (4x16 f32): lanes 0-15 col=lane K={0,1}; lanes 16-31 K={2,3}.
// C/D: VGPR r -> M = r + 8*hi, N = lane&15.
template <int KD, int ND>
__global__ __launch_bounds__(256) void k_gemm_wmma(const float* __restrict__ A,
                                                   const float* __restrict__ B,
                                                   float* __restrict__ C, int nrows) {
  constexpr int CT = ND / 16;     // column tiles covered by the 8 waves
  constexpr int RT = 8 / CT;      // row tiles per block
  const int wave  = threadIdx.x >> 5;
  const int lane  = threadIdx.x & 31;
  const int laneM = lane & 15;
  const int hi    = lane >> 4;    // 0 or 1 -> selects K pair {0,1} vs {2,3}
  const int rowt  = wave / CT;
  const int colt  = wave % CT;
  const int row0  = blockIdx.x * (16 * RT) + rowt * 16;
  const int col0  = colt * 16;
  if (row0 >= nrows) return;      // wave-uniform branch; EXEC stays all-1s

  const float* arow = A + (size_t)(row0 + laneM) * KD + 2 * hi;
  const float* bcol = B + (size_t)(2 * hi) * ND + col0 + laneM;

  v8f acc = {};
#pragma unroll 4
  for (int k = 0; k < KD; k += 4) {
    v2f a = *(const v2f*)(arow + k);                 // 8B aligned: k,2*hi even
    v2f b;
    b.x = bcol[(size_t)k * ND];
    b.y = bcol[(size_t)k * ND + ND];
    acc = __builtin_amdgcn_wmma_f32_16x16x4_f32(false, a, false, b, (short)0, acc,
                                                false, false);
  }

  float* crow = C + (size_t)(row0 + 8 * hi) * ND + col0 + laneM;
#pragma unroll
  for (int r = 0; r < 8; ++r) crow[(size_t)r * ND] = acc[r];
}

// ---------------- edge scatter: out[dst] += in[src] * dinv[s]*dinv[d] ------
template <int F>
__global__ __launch_bounds__(256) void k_edge_agg(const int* __restrict__ src,
                                                  const int* __restrict__ dst,
                                                  const float* __restrict__ dinv,
                                                  const float* __restrict__ xin,
                                                  float* __restrict__ out, int nE) {
  constexpr int EPB = 256 / F;
  int e = blockIdx.x * EPB + (threadIdx.x / F);
  int f = threadIdx.x & (F - 1);
  if (e >= nE) return;
  int s = src[e], d = dst[e];
  float nrm = dinv[s] * dinv[d];
  atomicAdd(out + (size_t)d * F + f, xin[(size_t)s * F + f] * nrm);
}

// ---------------- post layer 1: h = relu(agg + xw*dinv^2 + b1) -------------
__global__ void k_post1(float* __restrict__ h, const float* __restrict__ xw,
                        const float* __restrict__ b, const float* __restrict__ dinv,
                        int n) {
  int i = blockIdx.x * blockDim.x + threadIdx.x;
  int st = gridDim.x * blockDim.x;
  for (; i < n; i += st) {
    int node = i >> 7;            // F1 = 128
    int f = i & 127;
    float dv = dinv[node];
    float v = h[i] + xw[i] * dv * dv + b[f];
    h[i] = v > 0.0f ? v : 0.0f;
  }
}

// ---------------- post layer 2 + graph pooling (sum & count) ---------------
__global__ __launch_bounds__(256) void k_post2_pool(const float* __restrict__ agg2,
                                                    const float* __restrict__ h2,
                                                    const float* __restrict__ b2,
                                                    const float* __restrict__ dinv,
                                                    const int* __restrict__ batch,
                                                    float* __restrict__ gsum,
                                                    float* __restrict__ gcnt, int nN) {
  int node = blockIdx.x * 4 + (threadIdx.x >> 6);   // F2 = 64 -> 4 nodes/block
  int f = threadIdx.x & 63;
  if (node >= nN) return;
  float dv = dinv[node];
  size_t idx = (size_t)node * 64 + f;
  float v = agg2[idx] + h2[idx] * dv * dv + b2[f];
  int g = batch[node];
  atomicAdd(gsum + (size_t)g * 64 + f, v);
  if (f == 0) atomicAdd(gcnt + g, 1.0f);
}

// ---------------- head: mean, fc, log_softmax ------------------------------
__global__ void k_head(const float* __restrict__ gsum, const float* __restrict__ gcnt,
                       const float* __restrict__ fcW, const float* __restrict__ fcb,
                       float* __restrict__ out, int nG) {
  int g = blockIdx.x * blockDim.x + threadIdx.x;
  if (g >= nG) return;
  float inv = 1.0f / fmaxf(gcnt[g], 1.0f);
  float logit[4];
#pragma unroll
  for (int c = 0; c < 4; ++c) logit[c] = fcb[c];
  for (int f = 0; f < 64; ++f) {
    float m = gsum[(size_t)g * 64 + f] * inv;
#pragma unroll
    for (int c = 0; c < 4; ++c) logit[c] += m * fcW[f * 4 + c];
  }
  float mx = fmaxf(fmaxf(logit[0], logit[1]), fmaxf(logit[2], logit[3]));
  float s = 0.0f;
#pragma unroll
  for (int c = 0; c < 4; ++c) s += expf(logit[c] - mx);
  float lse = mx + logf(s);
#pragma unroll
  for (int c = 0; c < 4; ++c) out[(size_t)g * 4 + c] = logit[c] - lse;
}

// ---------------------------------------------------------------------------
extern "C" void kernel_launch(void* const* d_in, const int* in_sizes, int n_in,
                              void* d_out, int out_size, void* d_ws, size_t ws_size,
                              hipStream_t stream) {
  const float* x     = (const float*)d_in[0];
  const int*   ei    = (const int*)d_in[1];
  const int*   batch = (const int*)d_in[2];
  const float* W1    = (const float*)d_in[3];
  const float* b1    = (const float*)d_in[4];
  const float* W2    = (const float*)d_in[5];
  const float* b2    = (const float*)d_in[6];
  const float* fcW   = (const float*)d_in[7];
  const float* fcb   = (const float*)d_in[8];
  float* out = (float*)d_out;

  const int N = in_sizes[0] / 256;   // 100000 nodes
  const int E = in_sizes[1] / 2;     // 1600000 edges
  const int G = out_size / 4;        // 1000 graphs
  const int* src = ei;               // edge_index[0]
  const int* dst = ei + E;           // edge_index[1]

  char* ws = (char*)d_ws;
  size_t off = 0;
  auto alloc = [&](size_t nf) {
    float* p = (float*)(ws + off);
    off = (off + nf * sizeof(float) + 255) & ~(size_t)255;
    return p;
  };
  float* xw1  = alloc((size_t)N * 128);   // x @ W1
  float* h1   = alloc((size_t)N * 128);   // agg1, then relu'd layer-1 output
  float* h2   = alloc((size_t)N * 64);    // h1 @ W2
  float* agg2 = alloc((size_t)N * 64);
  float* dinv = alloc((size_t)N);         // degree, then rsqrt(degree)
  float* gsum = alloc((size_t)G * 64);
  float* gcnt = alloc((size_t)G);

  // --- init (re-done every call; buffers are not preserved between replays)
  k_fill<<<1024, 256, 0, stream>>>(dinv, 1.0f, (size_t)N);        // self-loop deg
  k_fill<<<4096, 256, 0, stream>>>(h1, 0.0f, (size_t)N * 128);
  k_fill<<<4096, 256, 0, stream>>>(agg2, 0.0f, (size_t)N * 64);
  k_fill<<<256, 256, 0, stream>>>(gsum, 0.0f, (size_t)G * 64);
  k_fill<<<8, 256, 0, stream>>>(gcnt, 0.0f, (size_t)G);

  // --- normalization
  k_degree<<<(E + 255) / 256, 256, 0, stream>>>(dst, dinv, E);
  k_rsqrt<<<(N + 255) / 256, 256, 0, stream>>>(dinv, N);

  // --- layer 1: GEMM (WMMA f32) -> edge scatter -> bias/self-loop/relu
  k_gemm_wmma<256, 128><<<(N + 15) / 16, 256, 0, stream>>>(x, W1, xw1, N);
  k_edge_agg<128><<<(E + 1) / 2, 256, 0, stream>>>(src, dst, dinv, xw1, h1, E);
  k_post1<<<4096, 256, 0, stream>>>(h1, xw1, b1, dinv, N * 128);

  // --- layer 2: GEMM (WMMA f32) -> edge scatter -> bias/self-loop + pooling
  k_gemm_wmma<128, 64><<<(N + 31) / 32, 256, 0, stream>>>(h1, W2, h2, N);
  k_edge_agg<64><<<(E + 3) / 4, 256, 0, stream>>>(src, dst, dinv, h2, agg2, E);
  k_post2_pool<<<(N + 3) / 4, 256, 0, stream>>>(agg2, h2, b2, dinv, batch, gsum, gcnt, N);

  // --- head
  k_head<<<(G + 255) / 256, 256, 0, stream>>>(gsum, gcnt, fcW, fcb, out, G);
}